// Net6_29755533427164
// MI455X (gfx1250) — compile-verified
//
#include <hip/hip_runtime.h>

#define N_NODES 100000
#define N_EDGES 3200000
#define D 16
#define L_LAYERS 3

typedef __attribute__((ext_vector_type(2))) float v2f;
typedef __attribute__((ext_vector_type(8))) float v8f;

__global__ void zero_f32(float* __restrict__ p, int n) {
  int i = blockIdx.x * blockDim.x + threadIdx.x;
  if (i < n) p[i] = 0.0f;
}

// in-degree (mean-aggregation denominator), computed once
__global__ void degree_kernel(const long long* __restrict__ ei,
                              float* __restrict__ deg) {
  int e = blockIdx.x * blockDim.x + threadIdx.x;
  if (e < N_EDGES) {
    int t = (int)ei[N_EDGES + e];
    atomicAdd(&deg[t], 1.0f);
  }
}

// scatter-add: 16 lanes per edge -> coalesced 64B gather of x[src] row,
// atomic f32 adds land in L2 (agg is 6.4MB, L2-resident)
__global__ void scatter_kernel(const long long* __restrict__ ei,
                               const float* __restrict__ x,
                               float* __restrict__ agg) {
  long long tid = (long long)blockIdx.x * blockDim.x + threadIdx.x;
  if (tid >= (long long)N_EDGES * D) return;
  int e = (int)(tid >> 4);
  int d = (int)(tid & 15);
  int s = (int)ei[e];
  int t = (int)ei[N_EDGES + e];
  atomicAdd(&agg[t * D + d], x[s * D + d]);
}

// Per-wave 16-node tile:  out = (agg/denom) @ Wl^T + x @ (Wr+Wlin)^T + (bl+blin)
// Chained V_WMMA_F32_16X16X4_F32, K=16 split into 4 chunks per matrix.
__global__ void node_kernel(const float* __restrict__ x,
                            const float* __restrict__ agg,
                            const float* __restrict__ deg,
                            const float* __restrict__ Wl,
                            const float* __restrict__ bl,
                            const float* __restrict__ Wr,
                            const float* __restrict__ Wlin,
                            const float* __restrict__ blin,
                            float* __restrict__ out) {
  const int lane = threadIdx.x & 31;
  const int wave = threadIdx.x >> 5;
  const int tile = blockIdx.x * (blockDim.x >> 5) + wave;
  const int NT = N_NODES / D;  // 6250 full tiles
  if (tile >= NT) return;      // wave-uniform: EXEC stays all-1s for WMMA
  const int m0 = tile * D;

  const int row = lane & 15;   // A-row M / B-col N / D-col N for this lane
  const int hi  = lane >> 4;   // K-half selector per ISA 32-bit A/B layout

  float dg  = deg[m0 + row];
  float inv = 1.0f / fmaxf(dg, 1.0f);

  v8f acc = {};  // C starts at zero

  // ---- acc += (agg * inv) @ Wl^T
  const float* arow = agg + (m0 + row) * D;
  #pragma unroll
  for (int kb = 0; kb < 4; ++kb) {
    int k0 = kb * 4 + hi * 2;                    // A VGPR0/1 hold K=k0,k0+1
    v2f a, b;
    a.x = arow[k0] * inv;
    a.y = arow[k0 + 1] * inv;
    b.x = Wl[row * D + k0];                      // B[k][N] = Wl[N][k]
    b.y = Wl[row * D + k0 + 1];
    acc = __builtin_amdgcn_wmma_f32_16x16x4_f32(
        false, a, false, b, (short)0, acc, false, false);
  }

  // ---- acc += x @ (Wr + Wlin)^T   (fused residual + conv right branch)
  const float* xrow = x + (m0 + row) * D;
  #pragma unroll
  for (int kb = 0; kb < 4; ++kb) {
    int k0 = kb * 4 + hi * 2;
    v2f a, b;
    a.x = xrow[k0];
    a.y = xrow[k0 + 1];
    b.x = Wr[row * D + k0]     + Wlin[row * D + k0];
    b.y = Wr[row * D + k0 + 1] + Wlin[row * D + k0 + 1];
    acc = __builtin_amdgcn_wmma_f32_16x16x4_f32(
        false, a, false, b, (short)0, acc, false, false);
  }

  // D layout: VGPR v holds M = v + 8*hi, N = row
  float bias = bl[row] + blin[row];
  #pragma unroll
  for (int v = 0; v < 8; ++v) {
    int M = v + hi * 8;
    out[(m0 + M) * D + row] = acc[v] + bias;
  }
}

extern "C" void kernel_launch(void* const* d_in, const int* in_sizes, int n_in,
                              void* d_out, int out_size, void* d_ws, size_t ws_size,
                              hipStream_t stream) {
  const float*     x0   = (const float*)d_in[0];
  const long long* ei   = (const long long*)d_in[1];  // int64 (2, N_EDGES)
  const float*     Wl   = (const float*)d_in[2];      // (L,16,16)
  const float*     bl   = (const float*)d_in[3];      // (L,16)
  const float*     Wr   = (const float*)d_in[4];
  const float*     Wlin = (const float*)d_in[5];
  const float*     blin = (const float*)d_in[6];
  float* out = (float*)d_out;

  float* deg = (float*)d_ws;                 // N_NODES
  float* agg = deg + N_NODES;                // N_NODES*D
  float* buf = agg + (size_t)N_NODES * D;    // N_NODES*D ping buffer

  zero_f32<<<(N_NODES + 255) / 256, 256, 0, stream>>>(deg, N_NODES);
  degree_kernel<<<(N_EDGES + 255) / 256, 256, 0, stream>>>(ei, deg);

  const long long ne16 = (long long)N_EDGES * D;
  const int scatter_blocks = (int)((ne16 + 255) / 256);
  const int node_blocks = (N_NODES / D + 7) / 8;  // 8 waves (tiles) per block

  const float* xin = x0;
  for (int l = 0; l < L_LAYERS; ++l) {
    float* xout = (l == L_LAYERS - 1) ? out : buf;
    zero_f32<<<(N_NODES * D + 255) / 256, 256, 0, stream>>>(agg, N_NODES * D);
    scatter_kernel<<<scatter_blocks, 256, 0, stream>>>(ei, xin, agg);
    node_kernel<<<node_blocks, 256, 0, stream>>>(
        xin, agg, deg,
        Wl + l * D * D, bl + l * D,
        Wr + l * D * D, Wlin + l * D * D, blin + l * D,
        xout);
    xin = xout;  // layer2 runs buf->buf in-place: safe, each wave reads
                 // only its own tile's rows before writing them
  }
}